// StructureTensorInferer_81561428951512
// MI455X (gfx1250) — compile-verified
//
#include <hip/hip_runtime.h>

// ---------------------------------------------------------------------------
// Separable structure-tensor pipeline for gfx1250 (MI455X).
// Each 1-D convolution pass runs as banded 16x16 GEMM tiles on
// V_WMMA_F32_16X16X4_F32 (fp32-exact). One wave owns 4 conv-axis tiles and
// reuses its banded-weight fragments across them. All boundary handling is
// branchless (clamped address + cndmask) so EXEC stays all-ones for WMMA.
// ---------------------------------------------------------------------------

typedef __attribute__((ext_vector_type(2))) float v2f;
typedef __attribute__((ext_vector_type(4))) float v4f;
typedef __attribute__((ext_vector_type(8))) float v8f;

namespace {

constexpr int DVOL = 192;
constexpr int SYs  = DVOL;
constexpr int SZs  = DVOL * DVOL;
constexpr int VOL  = DVOL * DVOL * DVOL;
constexpr int TILE = 16;
constexpr int TPB  = 256;                 // 8 wave32 per block
constexpr int WPB  = TPB / 32;
constexpr int TCW  = 4;                   // conv-axis tiles per wave (12 % 4 == 0)
constexpr int NTC  = (DVOL / TILE) / TCW; // 3 conv-tile groups
constexpr int PASS_WAVES  = NTC * (DVOL / TILE) * DVOL; // 6912
constexpr int PASS_BLOCKS = PASS_WAVES / WPB;           // 864 (exact)

// WSEL: 0 = 7-tap Gaussian (sigma=1), 1 = 5-tap s/16, 2 = 9-tap deriv d/96
template <int WSEL> __host__ __device__ constexpr int wtaps() {
  return WSEL == 0 ? 7 : (WSEL == 1 ? 5 : 9);
}

template <int WSEL>
__device__ __forceinline__ float wval(int k) {
  if (WSEL == 0) {
    const float g[7] = {0.00443305f, 0.05400558f, 0.24203623f, 0.39905028f,
                        0.24203623f, 0.05400558f, 0.00443305f};
    return g[k];
  } else if (WSEL == 1) {
    const float s[5] = {0.0625f, 0.25f, 0.375f, 0.25f, 0.0625f};
    return s[k];
  } else {
    const float d[9] = {0.020833333f, 0.010416667f, -0.166666667f, -0.28125f,
                        0.0f, 0.28125f, 0.166666667f, -0.010416667f, -0.020833333f};
    return d[k];
  }
}

// Banded weight: w[t] if 0<=t<TAPS else 0. Register-only select chain,
// computed once per wave per K-chunk (hoisted out of the tile loop).
template <int WSEL>
__device__ __forceinline__ float bandw(int t) {
  constexpr int T = wtaps<WSEL>();
  float r = 0.0f;
#pragma unroll
  for (int k = 0; k < T; ++k) r = (t == k) ? wval<WSEL>(k) : r;
  return r;
}

// AXIS: 0 = conv along x (data in A, coalesced stores; lines = y)
//       1 = conv along y (data in B; lines = x -> coalesced loads+stores)
//       2 = conv along z (data in B; lines = x -> coalesced loads+stores)
template <int WSEL, int AXIS>
__global__ __launch_bounds__(TPB) void conv1d_wmma(const float* __restrict__ src,
                                                   float* __restrict__ dst) {
  constexpr bool DIA   = (AXIS == 0);
  constexpr int sConv  = (AXIS == 0) ? 1 : (AXIS == 1 ? SYs : SZs);
  constexpr int sLine  = (AXIS == 0) ? SYs : 1;
  constexpr int sSlice = (AXIS == 2) ? SYs : SZs;
  constexpr int TAPS   = wtaps<WSEL>();
  constexpr int R      = (TAPS - 1) / 2;
  constexpr int NC     = (TILE + TAPS - 1 + 3) / 4;   // K chunks of 4

  const int lane = threadIdx.x & 31;
  const int wid  = blockIdx.x * WPB + (threadIdx.x >> 5);

  const int tg = wid % NTC;                       // conv-axis tile group (x4)
  const int tl = (wid / NTC) % (DVOL / TILE);     // line-axis tile
  const int sl = wid / (NTC * (DVOL / TILE));     // slice index

  const int half = lane >> 4;   // K pair select: {0,1} or {2,3}
  const int lid  = lane & 15;   // M (A rows) / N (B cols)
  const int k0   = half * 2;

  // Banded weight fragments: reused across all TCW tiles of this wave.
  v2f wf[NC];
#pragma unroll
  for (int c = 0; c < NC; ++c) {
    const int j0 = 4 * c + k0;
    wf[c].x = bandw<WSEL>(j0 - lid);
    wf[c].y = bandw<WSEL>(j0 + 1 - lid);
  }

  const int lineBase = sl * sSlice + (tl * TILE) * sLine + lid * sLine;
  const int outBase  = sl * sSlice + (tl * TILE) * sLine;

#pragma unroll 1
  for (int t = 0; t < TCW; ++t) {
    const int tc    = tg * TCW + t;
    const int conv0 = tc * TILE - R;   // conv-axis coordinate of K index j=0

    v8f acc = {};
#pragma unroll
    for (int c = 0; c < NC; ++c) {
      const int j0 = 4 * c + k0;
      // Branchless zero-padded loads: clamp address, cndmask the value.
      const int i0 = conv0 + j0;
      const int i1 = i0 + 1;
      const int c0 = min(DVOL - 1, max(0, i0));
      const int c1 = min(DVOL - 1, max(0, i1));
      const float v0 = src[lineBase + c0 * sConv];
      const float v1 = src[lineBase + c1 * sConv];
      v2f df;
      df.x = (i0 == c0) ? v0 : 0.0f;
      df.y = (i1 == c1) ? v1 : 0.0f;

      if (DIA) {
        acc = __builtin_amdgcn_wmma_f32_16x16x4_f32(false, df, false, wf[c],
                                                    (short)0, acc, false, false);
      } else {
        acc = __builtin_amdgcn_wmma_f32_16x16x4_f32(false, wf[c], false, df,
                                                    (short)0, acc, false, false);
      }
    }

    // D layout: VGPR i -> M = i + 8*half, N = lid.
#pragma unroll
    for (int i = 0; i < 8; ++i) {
      const int mo = i + half * 8;
      int addr;
      if (DIA) {   // m = line, n = conv-out (contiguous over lanes: x)
        addr = outBase + mo * sLine + (tc * TILE + lid) * sConv;
      } else {     // m = conv-out, n = line (contiguous over lanes: x)
        addr = outBase + lid * sLine + (tc * TILE + mo) * sConv;
      }
      dst[addr] = acc[i];
    }
  }
}

// Structure-tensor products, order [gz*gz, gy*gz, gx*gz, gy*gy, gx*gy, gx*gx].
__global__ __launch_bounds__(TPB) void products_kernel(
    const v4f* __restrict__ gz, const v4f* __restrict__ gy,
    const v4f* __restrict__ gx, v4f* __restrict__ j0, v4f* __restrict__ j1,
    v4f* __restrict__ j2, v4f* __restrict__ j3, v4f* __restrict__ j4,
    v4f* __restrict__ j5) {
  const int i = blockIdx.x * blockDim.x + threadIdx.x;
  const v4f a = gz[i], b = gy[i], c = gx[i];
  j0[i] = a * a; j1[i] = b * a; j2[i] = c * a;
  j3[i] = b * b; j4[i] = c * b; j5[i] = c * c;
}

template <int WSEL, int AXIS>
void launch_pass(hipStream_t st, const float* src, float* dst) {
  conv1d_wmma<WSEL, AXIS><<<PASS_BLOCKS, TPB, 0, st>>>(src, dst);
}

void conv_pass(hipStream_t st, const float* src, float* dst, int axis, int wsel) {
  if (axis == 0) {
    if (wsel == 0)      launch_pass<0, 0>(st, src, dst);
    else if (wsel == 1) launch_pass<1, 0>(st, src, dst);
    else                launch_pass<2, 0>(st, src, dst);
  } else if (axis == 1) {
    if (wsel == 0)      launch_pass<0, 1>(st, src, dst);
    else if (wsel == 1) launch_pass<1, 1>(st, src, dst);
    else                launch_pass<2, 1>(st, src, dst);
  } else {
    if (wsel == 0)      launch_pass<0, 2>(st, src, dst);
    else if (wsel == 1) launch_pass<1, 2>(st, src, dst);
    else                launch_pass<2, 2>(st, src, dst);
  }
}

} // namespace

extern "C" void kernel_launch(void* const* d_in, const int* in_sizes, int n_in,
                              void* d_out, int out_size, void* d_ws, size_t ws_size,
                              hipStream_t stream) {
  (void)in_sizes; (void)n_in; (void)out_size; (void)ws_size;

  const float* X = (const float*)d_in[0];   // 192^3 fp32; 1-D factors hardcoded
  float* O  = (float*)d_out;
  float *O0 = O, *O1 = O + VOL, *O2 = O + 2 * VOL, *O3 = O + 3 * VOL,
        *O4 = O + 4 * VOL, *O5 = O + 5 * VOL;
  float* B  = (float*)d_ws;                 // needs 3 volumes = ~85 MB
  float *B0 = B, *B1 = B + VOL, *B2 = B + 2 * VOL;

  // 1) xs = Gauss7^3(x): separable z,y,x                     -> xs in O0
  conv_pass(stream, X,  O0, 2, 0);
  conv_pass(stream, O0, O1, 1, 0);
  conv_pass(stream, O1, O0, 0, 0);
  // 2) shared smoothers
  conv_pass(stream, O0, O1, 0, 1);   // p = s_x(xs)
  conv_pass(stream, O0, O2, 1, 1);   // q = s_y(xs)
  // 3)
  conv_pass(stream, O1, O3, 1, 1);   // u = s_y(p)
  conv_pass(stream, O1, O4, 2, 1);   // v = s_z(p)
  conv_pass(stream, O2, O5, 2, 1);   // w = s_z(q)
  // 4) gradients
  conv_pass(stream, O3, O1, 2, 2);   // gz = d_z(u)
  conv_pass(stream, O4, O2, 1, 2);   // gy = d_y(v)
  conv_pass(stream, O5, B0, 0, 2);   // gx = d_x(w)
  // 5) tensor products: J0->O0, J1->O3, J2->O4, J3->O5, J4->B1, J5->B2
  products_kernel<<<(VOL / 4) / TPB, TPB, 0, stream>>>(
      (const v4f*)O1, (const v4f*)O2, (const v4f*)B0, (v4f*)O0, (v4f*)O3,
      (v4f*)O4, (v4f*)O5, (v4f*)B1, (v4f*)B2);
  // 6) per-channel Gauss7^3 smoothing (z,y,x), landing in final channel slots
  conv_pass(stream, O0, B0, 2, 0); conv_pass(stream, B0, O2, 1, 0); conv_pass(stream, O2, O0, 0, 0); // ch0 Jzz
  conv_pass(stream, O3, B0, 2, 0); conv_pass(stream, B0, O2, 1, 0); conv_pass(stream, O2, O1, 0, 0); // ch1 Jzy
  conv_pass(stream, O4, B0, 2, 0); conv_pass(stream, B0, O3, 1, 0); conv_pass(stream, O3, O2, 0, 0); // ch2 Jzx
  conv_pass(stream, O5, B0, 2, 0); conv_pass(stream, B0, O4, 1, 0); conv_pass(stream, O4, O3, 0, 0); // ch3 Jyy
  conv_pass(stream, B1, B0, 2, 0); conv_pass(stream, B0, O5, 1, 0); conv_pass(stream, O5, O4, 0, 0); // ch4 Jyx
  conv_pass(stream, B2, B0, 2, 0); conv_pass(stream, B0, B1, 1, 0); conv_pass(stream, B1, O5, 0, 0); // ch5 Jxx
}